// AttentionNetwork_86577950753177
// MI455X (gfx1250) — compile-verified
//
#include <hip/hip_runtime.h>
#include <hip/hip_bf16.h>

typedef __attribute__((ext_vector_type(2))) float v2f;
typedef __attribute__((ext_vector_type(4))) float v4f;
typedef __attribute__((ext_vector_type(8))) float v8f;

#define NN 4096
#define DD 128
#define MM 4
#define ROWS 8

// ---------------------------------------------------------------------------
// Kernel 1: ab[i][0..3] = log2e * (f_meta[i,:] @ W[:128, :])
//           ab[i][4..7] = log2e * (f_meta[i,:] @ W[128:, :])
// One wave (32 lanes) per 16-row tile, V_WMMA_F32_16X16X4_F32, K swept 128/4.
// ISA layout (§7.12.2):
//   A 16x4 f32 : lane l holds A[M = l%16, K = v + 2*(l/16)] in VGPR v (v=0,1)
//   B 4x16 f32 : lane l holds B[K = v + 2*(l/16), N = l%16] in VGPR v (mirror)
//   C/D 16x16  : VGPR r, lane l holds C[M = r + 8*(l/16), N = l%16]
// Wc[k][n] = (n<4) ? W[k][n] : (n<8) ? W[k+128][n-4] : 0
// ---------------------------------------------------------------------------
__global__ __launch_bounds__(32) void ab_gemm_wmma(const float* __restrict__ f,
                                                   const float* __restrict__ W,
                                                   float* __restrict__ ab) {
  const int lane = threadIdx.x & 31;
  const int hf   = lane >> 4;      // 0 or 1 (lane half)
  const int ln   = lane & 15;      // M for A-frag, N for B-frag
  const int row0 = blockIdx.x * 16;

  v8f c = {};
  #pragma unroll 4
  for (int k0 = 0; k0 < DD; k0 += 4) {
    const int ka = k0 + 2 * hf;    // this lane's first K
    // A fragment (two consecutive K values of row ln)
    const float* fp = f + (size_t)(row0 + ln) * DD + ka;
    v2f a;
    a.x = fp[0];
    a.y = fp[1];
    // B fragment, divergence-free (addresses always in range; zero via select)
    const int n4 = ln & 3;
    const int kb = ka + ((ln & 4) ? DD : 0);
    float b0 = W[(size_t)kb * MM + n4];
    float b1 = W[(size_t)(kb + 1) * MM + n4];
    if (ln >= 8) { b0 = 0.0f; b1 = 0.0f; }
    v2f b;
    b.x = b0;
    b.y = b1;
    // EXEC is all-ones here (no divergent control flow above)
    c = __builtin_amdgcn_wmma_f32_16x16x4_f32(false, a, false, b,
                                              (short)0, c, false, false);
  }

  const float LOG2E = 1.44269504088896340736f;
  if (ln < 8) {
    #pragma unroll
    for (int r = 0; r < 8; ++r) {
      ab[(size_t)(row0 + r + 8 * hf) * 8 + ln] = c[r] * LOG2E;
    }
  }
}

// ---------------------------------------------------------------------------
// Kernel 2: out[i][j][m] = e_m / sum_m(e_m),  e_m = exp2(max(ab[i][m]+ab[j][4+m],0))
// (ab already carries the log2e scale, so exp2 == exp of the original sum.)
// One thread per column j, ROWS rows per block: b-frag loaded once per thread,
// a-frag is block-uniform (scalar loads), one NT float4 store per (i,j).
// ---------------------------------------------------------------------------
__global__ __launch_bounds__(256) void pairwise_softmax(const float* __restrict__ ab,
                                                        float* __restrict__ out) {
  const int j  = blockIdx.x * 256 + threadIdx.x;
  const int i0 = blockIdx.y * ROWS;

  const v4f bb = *reinterpret_cast<const v4f*>(ab + (size_t)j * 8 + 4);

  #pragma unroll
  for (int r = 0; r < ROWS; ++r) {
    const int i = i0 + r;
    const v4f aa = *reinterpret_cast<const v4f*>(ab + (size_t)i * 8);  // uniform -> s_load

    const float t0 = fmaxf(aa.x + bb.x, 0.0f);
    const float t1 = fmaxf(aa.y + bb.y, 0.0f);
    const float t2 = fmaxf(aa.z + bb.z, 0.0f);
    const float t3 = fmaxf(aa.w + bb.w, 0.0f);

    const float e0 = __builtin_amdgcn_exp2f(t0);
    const float e1 = __builtin_amdgcn_exp2f(t1);
    const float e2 = __builtin_amdgcn_exp2f(t2);
    const float e3 = __builtin_amdgcn_exp2f(t3);

    const float inv = __builtin_amdgcn_rcpf(e0 + e1 + e2 + e3);  // sum >= 4, safe

    v4f o;
    o.x = e0 * inv;
    o.y = e1 * inv;
    o.z = e2 * inv;
    o.w = e3 * inv;

    // 256 MB pure write stream with zero reuse: bypass-ish via non-temporal hint
    __builtin_nontemporal_store(o, reinterpret_cast<v4f*>(out + ((size_t)i * NN + j) * 4));
  }
}

extern "C" void kernel_launch(void* const* d_in, const int* in_sizes, int n_in,
                              void* d_out, int out_size, void* d_ws, size_t ws_size,
                              hipStream_t stream) {
  const float* f_meta = (const float*)d_in[0];   // [4096, 128] fp32
  const float* W      = (const float*)d_in[1];   // [256, 4]   fp32
  float* out          = (float*)d_out;           // [4096, 4096, 4] fp32
  float* ab           = (float*)d_ws;            // [4096, 8] fp32 = 128 KB scratch

  hipLaunchKernelGGL(ab_gemm_wmma, dim3(NN / 16), dim3(32), 0, stream,
                     f_meta, W, ab);
  hipLaunchKernelGGL(pairwise_softmax, dim3(NN / 256, NN / ROWS), dim3(256), 0, stream,
                     ab, out);
}